// NeedlemanWunschAligner_89129161326628
// MI455X (gfx1250) — compile-verified
//
#include <hip/hip_runtime.h>
#include <hip/hip_bf16.h>

#define NEG (-1.0e10f)
#define BB 8
#define NN 512
#define MM 512
#define DD 512

typedef __attribute__((ext_vector_type(2))) float v2f;
typedef __attribute__((ext_vector_type(8))) float v8f;

__device__ __forceinline__ float softplus_f(float x) {
    return (x > 20.0f) ? x : log1pf(__expf(x));
}
__device__ __forceinline__ float logsigmoid_f(float x) {
    return (x < -20.0f) ? x : -log1pf(__expf(-x));
}

struct Frags {
    v2f a0, a1;            // two 16-row A fragments (rows r0..r0+15, r0+16..r0+31)
    v2f b0, b1, b2, b3;    // four 16-col B fragments
};

// ---------------------------------------------------------------------------
// Fused batched GEMM + activation:  O = act(X (N x D) * Y^T (D x M))
// grid.z in [0,16): z<8 -> theta (softplus, zx/zy), z>=8 -> A (logsigmoid, gx/gy)
// Block: 256 threads = 8 waves arranged 2 (rows) x 4 (cols).
// Wave tile: 32 rows x 64 cols = 8 accumulators; 8 WMMA : 6 loads per K-step,
// software-pipelined (next fragments loaded before current WMMAs issue).
// ---------------------------------------------------------------------------
__global__ __launch_bounds__(256) void gemm_act_kernel(
    const float* __restrict__ zx, const float* __restrict__ zy,
    const float* __restrict__ gx, const float* __restrict__ gy,
    float* __restrict__ theta_out, float* __restrict__ a_out) {

    const int z     = blockIdx.z;
    const int batch = z & 7;
    const bool isA  = (z >= BB);

    const float* __restrict__ X = (isA ? gx : zx) + (size_t)batch * NN * DD;
    const float* __restrict__ Y = (isA ? gy : zy) + (size_t)batch * MM * DD;
    float* __restrict__ O = (isA ? a_out : theta_out) + (size_t)batch * NN * MM;

    const int lane = threadIdx.x & 31;
    const int w    = threadIdx.x >> 5;     // 0..7
    const int wr   = w & 1;                // row group 0..1
    const int wc   = w >> 1;               // col group 0..3
    const int half = lane >> 4;            // K-pair selector within fragment
    const int lr   = lane & 15;

    const int row0 = blockIdx.y * 64 + wr * 32;   // output row base (X rows)
    const int col0 = blockIdx.x * 256 + wc * 64;  // output col base (Y rows)

    const float* __restrict__ xr0 = X + (size_t)(row0 +  0 + lr) * DD + 2 * half;
    const float* __restrict__ xr1 = X + (size_t)(row0 + 16 + lr) * DD + 2 * half;
    const float* __restrict__ yr0 = Y + (size_t)(col0 +  0 + lr) * DD + 2 * half;
    const float* __restrict__ yr1 = Y + (size_t)(col0 + 16 + lr) * DD + 2 * half;
    const float* __restrict__ yr2 = Y + (size_t)(col0 + 32 + lr) * DD + 2 * half;
    const float* __restrict__ yr3 = Y + (size_t)(col0 + 48 + lr) * DD + 2 * half;

    v8f acc00 = {}, acc01 = {}, acc02 = {}, acc03 = {};
    v8f acc10 = {}, acc11 = {}, acc12 = {}, acc13 = {};

    Frags cur;
    cur.a0 = *(const v2f*)(xr0);
    cur.a1 = *(const v2f*)(xr1);
    cur.b0 = *(const v2f*)(yr0);
    cur.b1 = *(const v2f*)(yr1);
    cur.b2 = *(const v2f*)(yr2);
    cur.b3 = *(const v2f*)(yr3);

#pragma unroll 4
    for (int kk = 4; kk < DD; kk += 4) {
        // Prefetch next K-step fragments into fresh registers so the WMMAs
        // below overlap with these loads (partial loadcnt waits, not 0-drains).
        Frags nxt;
        nxt.a0 = *(const v2f*)(xr0 + kk);
        nxt.a1 = *(const v2f*)(xr1 + kk);
        nxt.b0 = *(const v2f*)(yr0 + kk);
        nxt.b1 = *(const v2f*)(yr1 + kk);
        nxt.b2 = *(const v2f*)(yr2 + kk);
        nxt.b3 = *(const v2f*)(yr3 + kk);

        acc00 = __builtin_amdgcn_wmma_f32_16x16x4_f32(false, cur.a0, false, cur.b0,
                                                      (short)0, acc00, false, false);
        acc01 = __builtin_amdgcn_wmma_f32_16x16x4_f32(false, cur.a0, false, cur.b1,
                                                      (short)0, acc01, false, false);
        acc02 = __builtin_amdgcn_wmma_f32_16x16x4_f32(false, cur.a0, false, cur.b2,
                                                      (short)0, acc02, false, false);
        acc03 = __builtin_amdgcn_wmma_f32_16x16x4_f32(false, cur.a0, false, cur.b3,
                                                      (short)0, acc03, false, false);
        acc10 = __builtin_amdgcn_wmma_f32_16x16x4_f32(false, cur.a1, false, cur.b0,
                                                      (short)0, acc10, false, false);
        acc11 = __builtin_amdgcn_wmma_f32_16x16x4_f32(false, cur.a1, false, cur.b1,
                                                      (short)0, acc11, false, false);
        acc12 = __builtin_amdgcn_wmma_f32_16x16x4_f32(false, cur.a1, false, cur.b2,
                                                      (short)0, acc12, false, false);
        acc13 = __builtin_amdgcn_wmma_f32_16x16x4_f32(false, cur.a1, false, cur.b3,
                                                      (short)0, acc13, false, false);
        cur = nxt;
    }
    // Drain the final K-step.
    acc00 = __builtin_amdgcn_wmma_f32_16x16x4_f32(false, cur.a0, false, cur.b0,
                                                  (short)0, acc00, false, false);
    acc01 = __builtin_amdgcn_wmma_f32_16x16x4_f32(false, cur.a0, false, cur.b1,
                                                  (short)0, acc01, false, false);
    acc02 = __builtin_amdgcn_wmma_f32_16x16x4_f32(false, cur.a0, false, cur.b2,
                                                  (short)0, acc02, false, false);
    acc03 = __builtin_amdgcn_wmma_f32_16x16x4_f32(false, cur.a0, false, cur.b3,
                                                  (short)0, acc03, false, false);
    acc10 = __builtin_amdgcn_wmma_f32_16x16x4_f32(false, cur.a1, false, cur.b0,
                                                  (short)0, acc10, false, false);
    acc11 = __builtin_amdgcn_wmma_f32_16x16x4_f32(false, cur.a1, false, cur.b1,
                                                  (short)0, acc11, false, false);
    acc12 = __builtin_amdgcn_wmma_f32_16x16x4_f32(false, cur.a1, false, cur.b2,
                                                  (short)0, acc12, false, false);
    acc13 = __builtin_amdgcn_wmma_f32_16x16x4_f32(false, cur.a1, false, cur.b3,
                                                  (short)0, acc13, false, false);

    // C/D layout: VGPR r -> row (r + 8*half), column lr within each 16-wide tile.
    v8f accs0[4] = {acc00, acc01, acc02, acc03};
    v8f accs1[4] = {acc10, acc11, acc12, acc13};
#pragma unroll
    for (int g = 0; g < 2; ++g) {
        const int orow = row0 + g * 16 + 8 * half;
#pragma unroll
        for (int r = 0; r < 8; ++r) {
            float* orow_p = O + (size_t)(orow + r) * MM + col0 + lr;
#pragma unroll
            for (int c = 0; c < 4; ++c) {
                float v = (g == 0) ? accs0[c][r] : accs1[c][r];
                v = isA ? logsigmoid_f(v) : softplus_f(v);
                orow_p[c * 16] = v;
            }
        }
    }
}

// ---------------------------------------------------------------------------
// Forward soft-NW: one workgroup per batch, thread t owns row i=t+1.
// Diagonals double-buffered in LDS; each thread streams its theta/A row.
// Stores interior V (N x M per batch) to workspace for the backward pass.
// ---------------------------------------------------------------------------
__global__ __launch_bounds__(512) void nw_forward_kernel(
    const float* __restrict__ theta, const float* __restrict__ A,
    float* __restrict__ V) {

    __shared__ float buf0[NN + 1];
    __shared__ float buf1[NN + 1];

    const int b = blockIdx.x;
    const int t = threadIdx.x;   // 0..511
    const int i = t + 1;         // 1..512

    // buf0 = diag k-2 (init k=0: V[0,0]=0, rest NEG); buf1 = diag k-1 (all NEG)
    buf0[i] = NEG;
    buf1[i] = NEG;
    if (t == 0) { buf0[0] = 0.0f; buf1[0] = NEG; }
    __syncthreads();

    const float* __restrict__ th_r = theta + ((size_t)b * NN + t) * MM;
    const float* __restrict__ a_r  = A     + ((size_t)b * NN + t) * MM;
    float* __restrict__ v_r        = V     + ((size_t)b * NN + t) * MM;

    float* bm2 = buf0;   // diag k-2
    float* bm1 = buf1;   // diag k-1
    float vprev = NEG;   // V[i, j-1] (this thread's previous value)

    for (int k = 2; k <= NN + MM; ++k) {
        const int j = k - i;
        const bool act = (j >= 1) & (j <= MM);

        const float th = act ? th_r[j - 1] : NEG;
        const float a  = act ? a_r[j - 1]  : 0.0f;
        const float vu = bm1[i - 1];                 // V[i-1, j]
        const float vd = bm2[i - 1];                 // V[i-1, j-1]
        const float x0 = a + vu;
        const float x1 = vd;
        const float x2 = a + vprev;                  // V[i, j-1]
        const float m  = fmaxf(fmaxf(x0, x1), x2);
        float v = th + m + __logf(__expf(x0 - m) + __expf(x1 - m) + __expf(x2 - m));
        v = act ? v : NEG;

        __syncthreads();                 // all reads of bm2 done
        bm2[i] = v;                      // diag k overwrites diag k-2
        if (t == 0) bm2[0] = NEG;        // row-0 boundary is NEG for k>=1
        __syncthreads();                 // writes visible

        vprev = v;
        float* tmp = bm2; bm2 = bm1; bm1 = tmp;

        if (act) v_r[j - 1] = v;
    }
}

// ---------------------------------------------------------------------------
// Backward soft-NW (alignment posterior): E[i,j] = sum over successors of
// E[succ] * exp(edge + V[i,j] - (V[succ]-theta[succ])).  aln = E on interior.
// ---------------------------------------------------------------------------
__global__ __launch_bounds__(512) void nw_backward_kernel(
    const float* __restrict__ theta, const float* __restrict__ A,
    const float* __restrict__ V, float* __restrict__ aln) {

    __shared__ float ebuf0[NN + 2];
    __shared__ float ebuf1[NN + 2];

    const int b = blockIdx.x;
    const int t = threadIdx.x;
    const int i = t + 1;

    ebuf0[i] = 0.0f;
    ebuf1[i] = 0.0f;
    if (t == 0) { ebuf0[0] = 0.0f; ebuf1[0] = 0.0f;
                  ebuf0[NN + 1] = 0.0f; ebuf1[NN + 1] = 0.0f; }
    __syncthreads();

    const float* __restrict__ th_r  = theta + ((size_t)b * NN + t) * MM;
    const float* __restrict__ a_r   = A     + ((size_t)b * NN + t) * MM;
    const float* __restrict__ v_r   = V     + ((size_t)b * NN + t) * MM;
    const float* __restrict__ th_r1 = th_r + MM;   // 0-idx row i (successor row)
    const float* __restrict__ a_r1  = a_r + MM;
    const float* __restrict__ v_r1  = v_r + MM;
    float* __restrict__ o_r         = aln + ((size_t)b * NN + t) * MM;

    float* em1 = ebuf0;   // diag k+1
    float* em2 = ebuf1;   // diag k+2

    for (int k = NN + MM; k >= 2; --k) {
        const int j = k - i;
        const bool act = (j >= 1) & (j <= MM);

        float e = 0.0f;
        if (act) {
            const float vij = v_r[j - 1];
            if (i == NN && j == MM) e = 1.0f;           // terminal cell seeds E
            if (i + 1 <= NN) {
                // up successor (i+1, j)
                const float wu = __expf(a_r1[j - 1] + vij - (v_r1[j - 1] - th_r1[j - 1]));
                e += em1[i + 1] * wu;
                if (j + 1 <= MM) {
                    // diagonal successor (i+1, j+1)
                    const float wd = __expf(vij - (v_r1[j] - th_r1[j]));
                    e += em2[i + 1] * wd;
                }
            }
            if (j + 1 <= MM) {
                // left successor (i, j+1)
                const float wl = __expf(a_r[j] + vij - (v_r[j] - th_r[j]));
                e += em1[i] * wl;
            }
        }

        __syncthreads();
        em2[i] = e;                       // diag k overwrites diag k+2
        __syncthreads();
        float* tmp = em2; em2 = em1; em1 = tmp;

        if (act) o_r[j - 1] = e;
    }
}

// ---------------------------------------------------------------------------
extern "C" void kernel_launch(void* const* d_in, const int* in_sizes, int n_in,
                              void* d_out, int out_size, void* d_ws, size_t ws_size,
                              hipStream_t stream) {
    const float* zx = (const float*)d_in[0];
    const float* zy = (const float*)d_in[1];
    const float* gx = (const float*)d_in[2];
    const float* gy = (const float*)d_in[3];

    const size_t mat = (size_t)BB * NN * MM;   // 8*512*512
    float* aln_out   = (float*)d_out;
    float* theta_out = (float*)d_out + mat;
    float* a_out     = (float*)d_out + 2 * mat;
    float* Vws       = (float*)d_ws;           // 8*512*512 f32 = 8.4 MB scratch

    // 1) theta and A via f32 WMMA GEMMs (+ fused activations)
    dim3 ggrid(MM / 256, NN / 64, 2 * BB);
    gemm_act_kernel<<<ggrid, 256, 0, stream>>>(zx, zy, gx, gy, theta_out, a_out);

    // 2) forward soft-NW, storing V for the decode pass
    nw_forward_kernel<<<BB, NN, 0, stream>>>(theta_out, a_out, Vws);

    // 3) backward sweep -> alignment posterior (d score / d theta)
    nw_backward_kernel<<<BB, NN, 0, stream>>>(theta_out, a_out, Vws, aln_out);
}